// TAGNodeReg_56642028699868
// MI455X (gfx1250) — compile-verified
//
#include <hip/hip_runtime.h>
#include <hip/hip_bf16.h>

typedef __attribute__((ext_vector_type(2))) float v2f;
typedef __attribute__((ext_vector_type(8))) float v8f;

// ---------------------------------------------------------------------------
// Elementwise / setup kernels (float4-vectorized streaming: b128 ops)
// ---------------------------------------------------------------------------

__global__ void k_zero4(float4* __restrict__ p, long long n4) {
  long long i = (long long)blockIdx.x * blockDim.x + threadIdx.x;
  if (i < n4) p[i] = make_float4(0.f, 0.f, 0.f, 0.f);
}

__global__ void k_scatter_deg(float* __restrict__ deg, const int* __restrict__ dst,
                              const float* __restrict__ w, int E) {
  int e = blockIdx.x * blockDim.x + threadIdx.x;
  if (e < E) unsafeAtomicAdd(&deg[dst[e]], w[e]);
}

__global__ void k_dinv(float* __restrict__ dinv, const float* __restrict__ deg, int n) {
  int i = blockIdx.x * blockDim.x + threadIdx.x;
  if (i < n) {
    float d = deg[i];
    dinv[i] = (d > 0.0f) ? rsqrtf(fmaxf(d, 1e-30f)) : 0.0f;
  }
}

__global__ void k_norm(float* __restrict__ norm, const float* __restrict__ dinv,
                       const int* __restrict__ src, const int* __restrict__ dst,
                       const float* __restrict__ w, int E) {
  int e = blockIdx.x * blockDim.x + threadIdx.x;
  if (e < E) norm[e] = dinv[src[e]] * w[e] * dinv[dst[e]];
}

// acc[n, 64] = broadcast bias; float4 groups: group index within row = i & 15
__global__ void k_bias_init4(float4* __restrict__ acc, const float4* __restrict__ b4,
                             long long n4) {
  long long i = (long long)blockIdx.x * blockDim.x + threadIdx.x;
  if (i < n4) acc[i] = b4[i & 15];
}

__global__ void k_leaky4(float4* __restrict__ p, long long n4) {
  long long i = (long long)blockIdx.x * blockDim.x + threadIdx.x;
  if (i < n4) {
    float4 v = p[i];
    v.x = (v.x >= 0.0f) ? v.x : 0.01f * v.x;
    v.y = (v.y >= 0.0f) ? v.y : 0.01f * v.y;
    v.z = (v.z >= 0.0f) ? v.z : 0.01f * v.z;
    v.w = (v.w >= 0.0f) ? v.w : 0.01f * v.w;
    p[i] = v;
  }
}

// ---------------------------------------------------------------------------
// Sparse propagation: out[dst, :] += norm[e] * h[src, :]   (out pre-zeroed)
// One thread per (edge, 4-feature group): one global_load_b128 gather +
// 4 hardware global_atomic_add_f32 scatters. h/out are L2-resident
// (<=25.6 MB each vs 192 MB global L2), so gathers and atomics are served
// by L2; only indices + norm stream. Index/norm loads are redundant only
// D/4 times per edge (vs D with scalar threads).
// ---------------------------------------------------------------------------

template <int D>
__global__ void k_scatter_prop(float* __restrict__ out, const float* __restrict__ h,
                               const int* __restrict__ src, const int* __restrict__ dst,
                               const float* __restrict__ norm, int E) {
  constexpr int G = D / 4;                    // float4 groups per row
  constexpr int SH = (G == 16) ? 4 : 2;       // log2(G)
  long long t = (long long)blockIdx.x * blockDim.x + threadIdx.x;
  if (t >= (long long)E * G) return;
  int e = (int)(t >> SH);
  int g = (int)(t & (G - 1));
  float w = norm[e];
  float4 v = *((const float4*)(h + (long long)src[e] * D) + g);
  float* op = out + (long long)dst[e] * D + g * 4;
  unsafeAtomicAdd(op + 0, w * v.x);
  unsafeAtomicAdd(op + 1, w * v.y);
  unsafeAtomicAdd(op + 2, w * v.z);
  unsafeAtomicAdd(op + 3, w * v.w);
}

// ---------------------------------------------------------------------------
// GEMM-accumulate with fp32 WMMA:  out[N,64] += H[N,KDIM] @ W[KDIM,64]
// One wave per 16-row tile; 4 WMMA n-tiles cover the 64 output columns;
// K stepped by 4 via V_WMMA_F32_16X16X4_F32 (full fp32 precision).
//
// Fragment layouts (ISA 7.12.2):
//   A 16x4 f32 : lanes 0-15 -> M=lane, v0=K0,v1=K1 ; lanes 16-31 -> K2,K3
//   B 4x16 f32 : lanes carry N ; v0/v1 carry K rows (low lanes K0/K1, high K2/K3)
//   C/D 16x16  : VGPR i -> M=i (lanes 0-15), M=8+i (lanes 16-31), N=lane&15
// ---------------------------------------------------------------------------

template <int KDIM>
__global__ void k_gemm_acc(const float* __restrict__ H, const float* __restrict__ W,
                           float* __restrict__ out, int nTiles) {
  const int lane = threadIdx.x & 31;
  const int tile = blockIdx.x * (blockDim.x >> 5) + (threadIdx.x >> 5);
  if (tile >= nTiles) return;  // wave-uniform: EXEC stays all-ones for WMMA

  const int row0 = tile << 4;
  const int r    = lane & 15;        // M index within tile
  const int kh   = (lane >> 4) << 1; // K offset 0 or 2 (upper half-lanes)
  const int n0   = lane & 15;        // N index within n-tile
  const int mOff = (lane >> 4) << 3; // C/D row offset 0 or 8

  const float* Arow = H + (long long)(row0 + r) * KDIM + kh;
  float* Crow = out + (long long)(row0 + mOff) * 64 + n0;

  v8f acc[4];
#pragma unroll
  for (int nt = 0; nt < 4; ++nt) {
#pragma unroll
    for (int i = 0; i < 8; ++i) acc[nt][i] = Crow[(long long)i * 64 + nt * 16];
  }

#pragma unroll
  for (int k0 = 0; k0 < KDIM; k0 += 4) {
    v2f a;
    a.x = Arow[k0 + 0];
    a.y = Arow[k0 + 1];
    const float* Wk = W + (long long)(k0 + kh) * 64 + n0;
#pragma unroll
    for (int nt = 0; nt < 4; ++nt) {
      v2f b;
      b.x = Wk[nt * 16];
      b.y = Wk[nt * 16 + 64];
      acc[nt] = __builtin_amdgcn_wmma_f32_16x16x4_f32(
          /*neg_a=*/false, a, /*neg_b=*/false, b,
          /*c_mod=*/(short)0, acc[nt], /*reuse_a=*/false, /*reuse_b=*/false);
    }
  }

#pragma unroll
  for (int nt = 0; nt < 4; ++nt) {
#pragma unroll
    for (int i = 0; i < 8; ++i) Crow[(long long)i * 64 + nt * 16] = acc[nt][i];
  }
}

// ---------------------------------------------------------------------------
// Output head: y[i] = dot(h[i,:64], Wout) + bout
// ---------------------------------------------------------------------------

__global__ void k_final(float* __restrict__ y, const float* __restrict__ h,
                        const float* __restrict__ Wout, const float* __restrict__ bout,
                        int n) {
  int i = blockIdx.x * blockDim.x + threadIdx.x;
  if (i >= n) return;
  const float4* hr = (const float4*)(h + (long long)i * 64);
  const float4* wr = (const float4*)Wout;
  float s = 0.0f;
#pragma unroll
  for (int j = 0; j < 16; ++j) {
    float4 a = hr[j];
    float4 w = wr[j];
    s += a.x * w.x + a.y * w.y + a.z * w.z + a.w * w.w;
  }
  y[i] = s + bout[0];
}

// ---------------------------------------------------------------------------
// Host orchestration
// ---------------------------------------------------------------------------

static inline int cdiv_ll(long long a, long long b) { return (int)((a + b - 1) / b); }

extern "C" void kernel_launch(void* const* d_in, const int* in_sizes, int n_in,
                              void* d_out, int out_size, void* d_ws, size_t ws_size,
                              hipStream_t stream) {
  const float* x    = (const float*)d_in[0];   // [N,16]
  const int*   ei   = (const int*)d_in[1];     // [2,E]
  const float* ew   = (const float*)d_in[2];   // [E]
  // d_in[3] = batch (unused)
  const float* W1   = (const float*)d_in[4];   // [5,16,64]
  const float* b1   = (const float*)d_in[5];   // [64]
  const float* W2   = (const float*)d_in[6];   // [2,5,64,64]
  const float* b2   = (const float*)d_in[7];   // [2,64]
  const float* Wout = (const float*)d_in[8];   // [64,1]
  const float* bout = (const float*)d_in[9];   // [1]

  const int N = in_sizes[0] / 16;
  const int E = in_sizes[2];
  const int* src = ei;
  const int* dst = ei + E;

  // Workspace carve-up (all fp32): 2N + E + 4*64N floats (~116 MB)
  float* ws   = (float*)d_ws;
  float* deg  = ws;  ws += N;
  float* dinv = ws;  ws += N;
  float* nrm  = ws;  ws += E;
  float* bufA = ws;  ws += (long long)N * 64;
  float* bufB = ws;  ws += (long long)N * 64;
  float* bufC = ws;  ws += (long long)N * 64;
  float* bufD = ws;  ws += (long long)N * 64;

  const int TB = 256;
  const long long N64_4 = (long long)N * 16;  // N*64 floats / 4
  const long long N16_4 = (long long)N * 4;   // N*16 floats / 4

  // --- gcn_norm ---
  k_zero4<<<cdiv_ll(N / 4, TB), TB, 0, stream>>>((float4*)deg, N / 4);
  k_scatter_deg<<<cdiv_ll(E, TB), TB, 0, stream>>>(deg, dst, ew, E);
  k_dinv<<<cdiv_ll(N, TB), TB, 0, stream>>>(dinv, deg, N);
  k_norm<<<cdiv_ll(E, TB), TB, 0, stream>>>(nrm, dinv, src, dst, ew, E);

  const int nTiles = (N + 15) / 16;
  const int gemmBlocks = cdiv_ll(nTiles, TB / 32);

  // --- Layer 1: [N,16] -> [N,64] ---
  float* acc = bufB;
  k_bias_init4<<<cdiv_ll(N64_4, TB), TB, 0, stream>>>((float4*)acc, (const float4*)b1,
                                                      N64_4);
  k_gemm_acc<16><<<gemmBlocks, TB, 0, stream>>>(x, W1, acc, nTiles);
  {
    const float* hprev = x;
    float* pb = bufC;
    float* po = bufD;
    for (int k = 1; k <= 4; ++k) {
      k_zero4<<<cdiv_ll(N16_4, TB), TB, 0, stream>>>((float4*)pb, N16_4);
      k_scatter_prop<16><<<cdiv_ll((long long)E * 4, TB), TB, 0, stream>>>(
          pb, hprev, src, dst, nrm, E);
      k_gemm_acc<16><<<gemmBlocks, TB, 0, stream>>>(pb, W1 + (long long)k * 16 * 64,
                                                    acc, nTiles);
      hprev = pb;
      float* t = pb; pb = po; po = t;
    }
  }
  k_leaky4<<<cdiv_ll(N64_4, TB), TB, 0, stream>>>((float4*)acc, N64_4);
  const float* h = acc;  // bufB

  // --- Layers 2..3: [N,64] -> [N,64] ---
  for (int L = 0; L < 2; ++L) {
    float* acc2 = (h == bufB) ? bufA : bufB;
    const float* Wl = W2 + (long long)L * 5 * 64 * 64;
    const float* bl = b2 + (long long)L * 64;

    k_bias_init4<<<cdiv_ll(N64_4, TB), TB, 0, stream>>>((float4*)acc2,
                                                        (const float4*)bl, N64_4);
    k_gemm_acc<64><<<gemmBlocks, TB, 0, stream>>>(h, Wl, acc2, nTiles);

    const float* hprev = h;
    float* pb = bufC;
    float* po = bufD;
    for (int k = 1; k <= 4; ++k) {
      k_zero4<<<cdiv_ll(N64_4, TB), TB, 0, stream>>>((float4*)pb, N64_4);
      k_scatter_prop<64><<<cdiv_ll((long long)E * 16, TB), TB, 0, stream>>>(
          pb, hprev, src, dst, nrm, E);
      k_gemm_acc<64><<<gemmBlocks, TB, 0, stream>>>(pb, Wl + (long long)k * 64 * 64,
                                                    acc2, nTiles);
      hprev = pb;
      float* t = pb; pb = po; po = t;
    }
    k_leaky4<<<cdiv_ll(N64_4, TB), TB, 0, stream>>>((float4*)acc2, N64_4);
    h = acc2;
  }

  // --- Output head ---
  k_final<<<cdiv_ll(N, TB), TB, 0, stream>>>((float*)d_out, h, Wout, bout, N);
}